// GraphConvV2_53687091200299
// MI455X (gfx1250) — compile-verified
//
#include <hip/hip_runtime.h>

// ---------------------------------------------------------------------------
// GraphConv as gather + GEMM:  rows = M*V = 131072, K = R*C = 576, N = 64
// f32 WMMA (v_wmma_f32_16x16x4_f32), A tile + full weight matrix staged in
// the 320KB CDNA5 LDS, async global->LDS staging (gfx1250 ASYNCcnt path).
// ---------------------------------------------------------------------------

typedef __attribute__((ext_vector_type(2))) float v2f;
typedef __attribute__((ext_vector_type(4))) float f32x4;
typedef __attribute__((ext_vector_type(8))) float v8f;
typedef __attribute__((ext_vector_type(4))) int   i32x4;

#define AS_GLOBAL __attribute__((address_space(1)))
#define AS_LOCAL  __attribute__((address_space(3)))

constexpr int Mb = 32;            // batches
constexpr int Vv = 4096;          // vertices
constexpr int Rr = 9;             // regions
constexpr int Cc = 64;            // channels
constexpr int Uu = 64;            // units (N)
constexpr int Kk = Rr * Cc;       // 576 (K)

constexpr int BLOCK_ROWS = 64;    // output rows per block
constexpr int ASTRIDE    = 580;   // padded A row stride (floats): 580%64=4 -> conflict-free
constexpr int BSTRIDE    = 72;    // padded B row stride (floats): 2*72%64=16 -> half-waves disjoint, 16B aligned
constexpr int LDS_BYTES  = (BLOCK_ROWS * ASTRIDE + Kk * BSTRIDE) * 4; // 314368 < 320KB

#if __has_builtin(__builtin_amdgcn_global_load_async_to_lds_b128)
#define HAVE_ASYNC_LDS 1
#else
#define HAVE_ASYNC_LDS 0
#endif

__device__ __forceinline__ void copy16_g2l(const float* __restrict__ g, float* l) {
#if HAVE_ASYNC_LDS
    __builtin_amdgcn_global_load_async_to_lds_b128(
        (AS_GLOBAL i32x4*)g, (AS_LOCAL i32x4*)l, 0, 0);
#else
    *(f32x4*)l = *(const f32x4*)g;     // global_load_b128 + ds_store_b128
#endif
}

__device__ __forceinline__ void wait_async_done() {
#if HAVE_ASYNC_LDS
#if __has_builtin(__builtin_amdgcn_s_wait_asynccnt)
    __builtin_amdgcn_s_wait_asynccnt(0);
#else
    asm volatile("s_wait_asynccnt 0" ::: "memory");
#endif
#endif
}

__global__ __launch_bounds__(256, 1)
void graphconv_wmma_f32(const float* __restrict__ nodes,          // (M,V,C)
                        const int*   __restrict__ nodes_indices,  // (E,2) -> (b, src_v)
                        const float* __restrict__ kernelW,        // (K, U)
                        const float* __restrict__ bias,           // (U)
                        float*       __restrict__ out) {          // (M,V,U)
    extern __shared__ float smem[];
    float* As = smem;                                 // [BLOCK_ROWS][ASTRIDE]
    float* Bs = smem + BLOCK_ROWS * ASTRIDE;          // [Kk][BSTRIDE]

    const int tid     = threadIdx.x;
    const int rowBase = blockIdx.x * BLOCK_ROWS;      // global output row base
    const int m       = rowBase / Vv;                 // uniform per block (4096 % 64 == 0)

    // ---- stage gathered A tile: 64 rows x 9 regions x 64 ch  (9216 float4) ----
    for (int idx = tid; idx < BLOCK_ROWS * Rr * (Cc / 4); idx += 256) {
        const int part = idx & 15;                    // which float4 of the 64-ch segment
        const int seg  = idx >> 4;                    // 0..575
        const int r    = seg % Rr;
        const int lr   = seg / Rr;                    // local row 0..63
        const int e    = (rowBase + lr) * Rr + r;     // edge id
        const int src  = nodes_indices[2 * e + 1];    // source vertex
        const float* g = nodes + ((size_t)(m * Vv + src)) * Cc + part * 4;
        float*       l = As + lr * ASTRIDE + r * Cc + part * 4;
        copy16_g2l(g, l);
    }

    // ---- stage full weight matrix: 576 x 64  (9216 float4) ----
    for (int idx = tid; idx < Kk * (Uu / 4); idx += 256) {
        const int col4 = idx & 15;
        const int row  = idx >> 4;
        const float* g = kernelW + (size_t)row * Uu + col4 * 4;
        float*       l = Bs + row * BSTRIDE + col4 * 4;
        copy16_g2l(g, l);
    }

    wait_async_done();
    __syncthreads();

    // ---- per-wave tile assignment ----
    const int lane    = tid & 31;
    const int w       = tid >> 5;          // 0..7
    const int colTile = w & 3;             // 4 column tiles of 16
    const int trow0   = (w >> 2) * 2;      // waves 0-3 -> row tiles 0,1; 4-7 -> 2,3
    const int l16     = lane & 15;
    const int khalf   = lane >> 4;         // 0 or 1

    const int n  = colTile * 16 + l16;     // output column
    const float bv = bias[n];

    v8f c0, c1;
#pragma unroll
    for (int j = 0; j < 8; ++j) { c0[j] = bv; c1[j] = bv; }

    // A frag (16x4 f32): lanes 0-15 rows, VGPR0 = K{+0|+2}, VGPR1 = K{+1|+3}
    const float* aRow0 = As + (trow0 * 16 + l16) * ASTRIDE + 2 * khalf;
    const float* aRow1 = aRow0 + 16 * ASTRIDE;
    // B frag (4x16 f32): lanes 0-15 cols, VGPR0 = K{+0|+2}, VGPR1 = K{+1|+3}
    const float* bCol  = Bs + 2 * khalf * BSTRIDE + n;

#pragma unroll 4
    for (int k = 0; k < Kk; k += 4) {
        v2f a0 = *(const v2f*)(aRow0 + k);            // ds_load_b64, conflict-free
        v2f a1 = *(const v2f*)(aRow1 + k);
        v2f b;
        b.x = bCol[(size_t)k * BSTRIDE];
        b.y = bCol[(size_t)(k + 1) * BSTRIDE];
        c0 = __builtin_amdgcn_wmma_f32_16x16x4_f32(false, a0, false, b,
                                                   (short)0, c0, false, false);
        c1 = __builtin_amdgcn_wmma_f32_16x16x4_f32(false, a1, false, b,
                                                   (short)0, c1, false, false);
    }

    // ---- fused ReLU + store (C/D layout: VGPR j -> M=j (lo half), M=j+8 (hi half)) ----
    float* outBase = out + (size_t)rowBase * Uu + n;
#pragma unroll
    for (int j = 0; j < 8; ++j) {
        const int r0 = trow0 * 16 + j + 8 * khalf;
        const int r1 = r0 + 16;
        const float x0 = c0[j] > 0.f ? c0[j] : 0.f;
        const float x1 = c1[j] > 0.f ? c1[j] : 0.f;
        outBase[(size_t)r0 * Uu] = x0;
        outBase[(size_t)r1 * Uu] = x1;
    }
}

extern "C" void kernel_launch(void* const* d_in, const int* in_sizes, int n_in,
                              void* d_out, int out_size, void* d_ws, size_t ws_size,
                              hipStream_t stream) {
    (void)in_sizes; (void)n_in; (void)out_size; (void)d_ws; (void)ws_size;
    const float* nodes         = (const float*)d_in[0];
    const int*   nodes_indices = (const int*)d_in[1];
    // d_in[2] = column_indices: deterministic (b,v,r) enumeration -> not needed
    const float* kernelW       = (const float*)d_in[3];
    const float* bias          = (const float*)d_in[4];
    float*       out           = (float*)d_out;

    static_assert(LDS_BYTES <= 320 * 1024, "LDS tile exceeds WGP LDS");

    // Allow >64KB dynamic LDS (capture-safe: not a stream op)
    (void)hipFuncSetAttribute((const void*)graphconv_wmma_f32,
                              hipFuncAttributeMaxDynamicSharedMemorySize, LDS_BYTES);

    const int nBlocks = (Mb * Vv) / BLOCK_ROWS;   // 2048
    graphconv_wmma_f32<<<nBlocks, 256, LDS_BYTES, stream>>>(
        nodes, nodes_indices, kernelW, bias, out);
}